// PointConvMessagePassing_34291018891266
// MI455X (gfx1250) — compile-verified
//
#include <hip/hip_runtime.h>
#include <math.h>

typedef __attribute__((ext_vector_type(2))) float v2f;
typedef __attribute__((ext_vector_type(8))) float v8f;

#define N_NODES 20000
#define N_EDGES 160000
#define MULC 64
#define EDIMC 8

__device__ __forceinline__ float silu_f(float x) { return x / (1.0f + __expf(-x)); }

// C += A(16xK, row stride lda, col step stepA) * B(Kx16, row stride ldb)
// f32 WMMA 16x16x4 fragment layout (ISA 7.12.2):
//   A: lane -> M=lane&15, half-wave selects K pair {0,1}/{2,3}; 2 VGPRs
//   B: lane -> N=lane&15, half-wave selects K pair; 2 VGPRs
//   C/D: lane -> N=lane&15, VGPR r -> M = r + 8*(lane>=16)
__device__ __forceinline__ void wmma_acc(const float* A, int lda, int stepA,
                                         const float* B, int ldb, int K, v8f& c) {
  const int lane = threadIdx.x & 31;
  const int m  = lane & 15;
  const int k0 = (lane >> 4) << 1;
  for (int k = 0; k < K; k += 4) {
    v2f a, b;
    a.x = A[m * lda + (k + k0) * stepA];
    a.y = A[m * lda + (k + k0 + 1) * stepA];
    b.x = B[(k + k0) * ldb + m];
    b.y = B[(k + k0 + 1) * ldb + m];
    c = __builtin_amdgcn_wmma_f32_16x16x4_f32(false, a, false, b, (short)0, c,
                                              false, false);
  }
}

// Contiguous-A variant (stepA==1): A pair is one 8B-aligned b64 load.
__device__ __forceinline__ void wmma_acc_c(const float* A, int lda,
                                           const float* B, int ldb, int K, v8f& c) {
  const int lane = threadIdx.x & 31;
  const int m  = lane & 15;
  const int k0 = (lane >> 4) << 1;
  for (int k = 0; k < K; k += 4) {
    v2f a = *(const v2f*)(A + m * lda + k + k0);
    v2f b;
    b.x = B[(k + k0) * ldb + m];
    b.y = B[(k + k0 + 1) * ldb + m];
    c = __builtin_amdgcn_wmma_f32_16x16x4_f32(false, a, false, b, (short)0, c,
                                              false, false);
  }
}

// C += kron(x, attrs)(16x256) * B(256x16).  A[m, 4u+z] = x[m,u]*attrs[m,z].
// K-pair per lane has constant z = {k0, k0+1}, so attrs are loop-invariant.
__device__ __forceinline__ void wmma_kron_acc(const float* x, int ldx, int stepx,
                                              const float* attrs,  // [m*4+z]
                                              const float* B, int ldb, v8f& c) {
  const int lane = threadIdx.x & 31;
  const int m  = lane & 15;
  const int k0 = (lane >> 4) << 1;
  const float za = attrs[m * 4 + k0];
  const float zb = attrs[m * 4 + k0 + 1];
  for (int t = 0; t < 64; ++t) {
    const float xv = x[m * ldx + t * stepx];
    v2f a; a.x = xv * za; a.y = xv * zb;
    const int k = 4 * t + k0;
    v2f b; b.x = B[k * ldb + m]; b.y = B[(k + 1) * ldb + m];
    c = __builtin_amdgcn_wmma_f32_16x16x4_f32(false, a, false, b, (short)0, c,
                                              false, false);
  }
}

__device__ __forceinline__ void store_tile1(v8f c, float s, float* D, int ldd, int stepD) {
  const int lane = threadIdx.x & 31;
  const int n  = lane & 15;
  const int mb = (lane >> 4) << 3;
#pragma unroll
  for (int r = 0; r < 8; ++r)
    D[(mb + r) * ldd + n * stepD] = c[r] * s;
}

__device__ __forceinline__ void store_tile2(v8f c1, float s1s, v8f c2, float s2s,
                                            float* D, int ldd, int stepD) {
  const int lane = threadIdx.x & 31;
  const int n  = lane & 15;
  const int mb = (lane >> 4) << 3;
#pragma unroll
  for (int r = 0; r < 8; ++r)
    D[(mb + r) * ldd + n * stepD] = c1[r] * s1s + c2[r] * s2s;
}

__global__ void __launch_bounds__(256) zero_kernel(float4* p, int n4) {
  int i = blockIdx.x * 256 + threadIdx.x;
  if (i < n4) p[i] = make_float4(0.f, 0.f, 0.f, 0.f);
}

// h0 = x0 @ W_lin1_0 / 8 ; h1[:,v,d] = sum_u x1[:,u,d] W_lin1_1[u,v] / 8
// h layout per node: [0:64]=h0, [64 + u*3 + d]=h1  (written into d_out scratch)
__global__ void __launch_bounds__(256) node_lin1_kernel(
    const float* __restrict__ nf, const float* __restrict__ Wl10,
    const float* __restrict__ Wl11, float* __restrict__ h) {
  const int nb = blockIdx.x * 16;
  const int t  = blockIdx.y * 8 + (threadIdx.x >> 5);  // 0..15, wave-uniform
  const float inv = 0.125f;                            // 1/sqrt(64)
  v8f c = {};
  if (t < 4) {
    wmma_acc_c(nf + (size_t)nb * 256, 256, Wl10 + t * 16, 64, 64, c);
    store_tile1(c, inv, h + (size_t)nb * 256 + t * 16, 256, 1);
  } else {
    const int jj = t - 4, vt = jj & 3, d = jj >> 2;
    wmma_acc(nf + (size_t)nb * 256 + 64 + d, 256, 3, Wl11 + vt * 16, 64, 64, c);
    store_tile1(c, inv, h + (size_t)nb * 256 + 64 + vt * 48 + d, 256, 3);
  }
}

// One wave (32 lanes) per edge: MLP -> w1..w4, gather h[src], messages, scatter.
__global__ void __launch_bounds__(256) edge_kernel(
    const float* __restrict__ h, const float* __restrict__ eattr,
    const float* __restrict__ eemb, const float* __restrict__ Wm1,
    const float* __restrict__ Wm2, const int* __restrict__ eidx,
    float* __restrict__ s0, float* __restrict__ s1) {
  __shared__ __align__(16) float w2s[EDIMC * 4 * MULC];  // 8 x 256 = 8KB

  // Stage W_mlp2 into LDS with direct memory->LDS async copies (ASYNCcnt),
  // bypassing the VGPR relay: 512 x b128 across the block.
  {
    const unsigned long long gbase = (unsigned long long)(uintptr_t)Wm2;
    const unsigned int lbase = (unsigned int)(uintptr_t)w2s;  // low 32 = LDS offset
    for (int i = threadIdx.x; i < (EDIMC * 4 * MULC) / 4; i += 256) {
      unsigned int la = lbase + (unsigned int)i * 16u;
      unsigned long long ga = gbase + (unsigned long long)i * 16ull;
      asm volatile("global_load_async_to_lds_b128 %0, %1, off"
                   :: "v"(la), "v"(ga) : "memory");
    }
    asm volatile("s_wait_asynccnt 0x0" ::: "memory");
  }
  __syncthreads();

  const int lane = threadIdx.x & 31;
  const int e = blockIdx.x * 8 + (threadIdx.x >> 5);

  const float inv_e = 0.35355339059327373f;  // 1/sqrt(8)
  const float inv_n = 0.35355339059327373f;  // 1/sqrt(8)
  const float inv_s3 = 0.5773502691896258f;  // 1/sqrt(3)

  const int src = eidx[e];
  const int dst = eidx[N_EDGES + e];

  // layer 1: lanes j=lane&7 each produce hid[j] cooperatively (wave32 shfl)
  const float embv = eemb[(size_t)e * EDIMC + (lane & 7)];
  const int j = lane & 7;
  float hsum = 0.0f;
#pragma unroll
  for (int i = 0; i < EDIMC; ++i) hsum += __shfl(embv, i) * Wm1[i * EDIMC + j];
  const float hidv = silu_f(hsum * inv_e);
  float hid[EDIMC];
#pragma unroll
  for (int i = 0; i < EDIMC; ++i) hid[i] = __shfl(hidv, i);

  const float a0 = eattr[(size_t)e * 4 + 0];
  float a1[3];
#pragma unroll
  for (int d = 0; d < 3; ++d) a1[d] = eattr[(size_t)e * 4 + 1 + d];

#pragma unroll
  for (int half = 0; half < 2; ++half) {
    const int u = lane + half * 32;  // channel 0..63
    float w[4];
#pragma unroll
    for (int g = 0; g < 4; ++g) {
      float acc = 0.0f;
#pragma unroll
      for (int i2 = 0; i2 < EDIMC; ++i2)
        acc += hid[i2] * w2s[i2 * 256 + g * MULC + u];
      w[g] = acc * inv_e;
    }
    const float g0 = h[(size_t)src * 256 + u];
    float g1[3];
#pragma unroll
    for (int d = 0; d < 3; ++d) g1[d] = h[(size_t)src * 256 + 64 + u * 3 + d];
    const float dot = g1[0] * a1[0] + g1[1] * a1[1] + g1[2] * a1[2];

    atomicAdd(&s0[(size_t)dst * 128 + u],      w[0] * g0 * a0 * inv_n);
    atomicAdd(&s0[(size_t)dst * 128 + 64 + u], w[3] * dot * inv_s3 * inv_n);
#pragma unroll
    for (int d = 0; d < 3; ++d) {
      atomicAdd(&s1[(size_t)dst * 384 + u * 3 + d],        w[1] * g0 * a1[d] * inv_n);
      atomicAdd(&s1[(size_t)dst * 384 + (64 + u) * 3 + d], w[2] * g1[d] * a0 * inv_n);
    }
  }
}

// t0 = s0@Wl20*inv2 + kron(x0,attr)@Wsc0*inv_sc ; t1 likewise per d-slice;
// out = nf + [silu(t0[:64]), (silu(t0[64:])[:,None]*t1).flat]
__global__ void __launch_bounds__(256) node_out_kernel(
    const float* __restrict__ nf, const float* __restrict__ na,
    const float* __restrict__ Wl20, const float* __restrict__ Wl21,
    const float* __restrict__ Wsc0, const float* __restrict__ Wsc1,
    const float* __restrict__ s0, const float* __restrict__ s1,
    float* __restrict__ out) {
  __shared__ float tbuf[16 * 320];  // per node: t0 [0:128], t1 [128 + u*3 + d]
  const int nb = blockIdx.x * 16;
  const int wave = threadIdx.x >> 5;
  const float inv2 = 0.08838834764831845f;  // 1/sqrt(128)
  const float inv_sc = 0.0625f;             // 1/sqrt(256)

  // Warm WGP-level cache for the (L2-resident) weight blocks.
  __builtin_prefetch(Wl20 + (threadIdx.x << 2), 0, 3);
  __builtin_prefetch(Wsc0 + (threadIdx.x << 2), 0, 3);

  for (int q = wave; q < 20; q += 8) {  // wave-uniform tiles
    v8f c1 = {}, c2 = {};
    if (q < 8) {
      wmma_acc_c(s0 + (size_t)nb * 128, 128, Wl20 + q * 16, 128, 128, c1);
      wmma_kron_acc(nf + (size_t)nb * 256, 256, 1, na + (size_t)nb * 4,
                    Wsc0 + q * 16, 128, c2);
      store_tile2(c1, inv2, c2, inv_sc, tbuf + q * 16, 320, 1);
    } else {
      const int jj = q - 8, vt = jj & 3, d = jj >> 2;
      wmma_acc(s1 + (size_t)nb * 384 + d, 384, 3, Wl21 + vt * 16, 64, 128, c1);
      wmma_kron_acc(nf + (size_t)nb * 256 + 64 + d, 256, 3, na + (size_t)nb * 4,
                    Wsc1 + vt * 16, 64, c2);
      store_tile2(c1, inv2, c2, inv_sc, tbuf + 128 + vt * 48 + d, 320, 3);
    }
  }
  __syncthreads();

  for (int i = threadIdx.x; i < 16 * 256; i += 256) {
    const int m = i >> 8;
    const int jc = i & 255;
    const float* t = tbuf + m * 320;
    float v;
    if (jc < 64) {
      v = silu_f(t[jc]);
    } else {
      const int u = (jc - 64) / 3;
      const int d = (jc - 64) % 3;
      v = silu_f(t[64 + u]) * t[128 + u * 3 + d];
    }
    out[(size_t)(nb + m) * 256 + jc] = nf[(size_t)(nb + m) * 256 + jc] + v;
  }
}

extern "C" void kernel_launch(void* const* d_in, const int* in_sizes, int n_in,
                              void* d_out, int out_size, void* d_ws, size_t ws_size,
                              hipStream_t stream) {
  const float* nf   = (const float*)d_in[0];
  const float* na   = (const float*)d_in[1];
  const float* ea   = (const float*)d_in[2];
  const float* ee   = (const float*)d_in[3];
  const float* Wl10 = (const float*)d_in[4];
  const float* Wl11 = (const float*)d_in[5];
  const float* Wm1  = (const float*)d_in[6];
  const float* Wm2  = (const float*)d_in[7];
  const float* Wl20 = (const float*)d_in[8];
  const float* Wl21 = (const float*)d_in[9];
  const float* Wsc0 = (const float*)d_in[10];
  const float* Wsc1 = (const float*)d_in[11];
  const int*   eidx = (const int*)d_in[12];
  float* out = (float*)d_out;

  float* s0 = (float*)d_ws;                       // N x 128
  float* s1 = s0 + (size_t)N_NODES * 128;         // N x 384

  const int ztotal4 = N_NODES * 512 / 4;
  zero_kernel<<<(ztotal4 + 255) / 256, 256, 0, stream>>>((float4*)d_ws, ztotal4);
  // h written into d_out as scratch (overwritten by node_out_kernel at the end)
  node_lin1_kernel<<<dim3(N_NODES / 16, 2), 256, 0, stream>>>(nf, Wl10, Wl11, out);
  edge_kernel<<<N_EDGES / 8, 256, 0, stream>>>(out, ea, ee, Wm1, Wm2, eidx, s0, s1);
  node_out_kernel<<<N_NODES / 16, 256, 0, stream>>>(nf, na, Wl20, Wl21, Wsc0, Wsc1,
                                                    s0, s1, out);
}